// VDP_intermediate_Conv_27745488732892
// MI455X (gfx1250) — compile-verified
//
#include <hip/hip_runtime.h>
#include <math.h>
#include <stdint.h>

typedef __attribute__((ext_vector_type(2))) float v2f;
typedef __attribute__((ext_vector_type(8))) float v8f;

#define KSZ   5
#define KNUM  32
#define CIN   16
#define HIMG  64
#define WIMG  64
#define BSZ   32
#define HOUT  60
#define WOUT  60
#define PNUM  (HOUT*WOUT)            // 3600 positions / batch
#define DK    (KSZ*KSZ*CIN)          // 400 patch length
#define OUTN  (BSZ*PNUM*KNUM)        // 3,686,400 per output tensor
#define TILES_PER_BATCH (PNUM/16)    // 225 (exact)
#define WAVES_PER_BLOCK 8
#define PRIOR_VAR 0.01f

__device__ __forceinline__ float softplusf(float x) {
    return (x > 20.0f) ? x : log1pf(__expf(x));
}

__global__ __launch_bounds__(256)
void vdp_conv_main(const float* __restrict__ mu_in,
                   const float* __restrict__ Sigma_in,
                   const float* __restrict__ w_mu,
                   const float* __restrict__ w_sigma,
                   float* __restrict__ out)
{
    // Weights staged to LDS in layout [(d>>2)][n][d&3] so each lane's B
    // operand (two consecutive K values for one column n) is a single b64 load.
    __shared__ float w_lds[DK * KNUM];   // 51.2 KB

    const int tid = threadIdx.x;
    // Async global->LDS staging (gfx1250 ASYNCcnt path); per-lane scatter
    // implements the (d,n) -> ((d>>2),n,(d&3)) permutation.
    for (int i = tid; i < DK * KNUM; i += 256) {
        int d = i >> 5;          // patch element
        int n = i & 31;          // output channel
        unsigned lds_off = (unsigned)(uintptr_t)(&w_lds[(d >> 2) * 128 + n * 4 + (d & 3)]);
        const float* gptr = w_mu + i;
        asm volatile("global_load_async_to_lds_b32 %0, %1, off"
                     :: "v"(lds_off), "v"(gptr) : "memory");
    }
    asm volatile("s_wait_asynccnt 0x0" ::: "memory");
    __syncthreads();

    const int lane   = tid & 31;
    const int wave   = tid >> 5;
    const int pairId = blockIdx.x * WAVES_PER_BLOCK + wave;   // 0..3599
    const int tA     = pairId * 2;                            // two 16-row tiles per wave
    const int tB     = tA + 1;

    const int laneHi = (lane >> 4) & 1;     // 0: K={0,1}, 1: K={2,3} of each 4-step
    const int laneK  = laneHi * 2;
    const int nlo    = lane & 15;
    const int M      = lane & 15;           // A-operand row (ISA 16x4 f32 A layout)

    const int bA = tA / TILES_PER_BATCH, pbA = (tA % TILES_PER_BATCH) * 16;
    const int bB = tB / TILES_PER_BATCH, pbB = (tB % TILES_PER_BATCH) * 16;
    const int pA = pbA + M, yA = pA / WOUT, xA = pA % WOUT;
    const int pB = pbB + M, yB = pB / WOUT, xB = pB % WOUT;

    const float* amuA = mu_in    + (((size_t)(bA * HIMG + yA) * WIMG + xA) * CIN + laneK);
    const float* asgA = Sigma_in + (((size_t)(bA * HIMG + yA) * WIMG + xA) * CIN + laneK);
    const float* amuB = mu_in    + (((size_t)(bB * HIMG + yB) * WIMG + xB) * CIN + laneK);
    const float* asgB = Sigma_in + (((size_t)(bB * HIMG + yB) * WIMG + xB) * CIN + laneK);

    const int lb0 = nlo * 4 + laneK;          // B tile 0 (n = 0..15)
    const int lb1 = (16 + nlo) * 4 + laneK;   // B tile 1 (n = 16..31)

    v8f accMuA0 = {}; v8f accMuA1 = {}; v8f accSgA0 = {}; v8f accSgA1 = {};
    v8f accMuB0 = {}; v8f accMuB1 = {}; v8f accSgB0 = {}; v8f accSgB1 = {};
    float rsA = 0.0f, rsB = 0.0f;             // combined trace + ||mu||^2 partials
    const float inv_d = 1.0f / (float)DK;

    #pragma unroll 1
    for (int i = 0; i < KSZ; ++i) {
        #pragma unroll
        for (int j = 0; j < KSZ; ++j) {
            const int eo = (i * WIMG + j) * CIN;
            const float* amA = amuA + eo;
            const float* asA = asgA + eo;
            const float* amB = amuB + eo;
            const float* asB = asgB + eo;
            const int dq = (i * KSZ + j) * 4;
            #pragma unroll
            for (int cc = 0; cc < 4; ++cc) {
                v2f aMuA = *(const v2f*)(amA + cc * 4);
                v2f aSgA = *(const v2f*)(asA + cc * 4);
                v2f aMuB = *(const v2f*)(amB + cc * 4);
                v2f aSgB = *(const v2f*)(asB + cc * 4);
                const float* wl = &w_lds[(dq + cc) * 128];
                v2f bw0 = *(const v2f*)(wl + lb0);
                v2f bw1 = *(const v2f*)(wl + lb1);
                v2f b20, b21;                       // w^2 / 400 on the fly, shared by both tiles
                b20.x = bw0.x * bw0.x * inv_d;  b20.y = bw0.y * bw0.y * inv_d;
                b21.x = bw1.x * bw1.x * inv_d;  b21.y = bw1.y * bw1.y * inv_d;

                accMuA0 = __builtin_amdgcn_wmma_f32_16x16x4_f32(false, aMuA, false, bw0,
                                                                (short)0, accMuA0, false, false);
                accMuA1 = __builtin_amdgcn_wmma_f32_16x16x4_f32(false, aMuA, false, bw1,
                                                                (short)0, accMuA1, false, false);
                accMuB0 = __builtin_amdgcn_wmma_f32_16x16x4_f32(false, aMuB, false, bw0,
                                                                (short)0, accMuB0, false, false);
                accMuB1 = __builtin_amdgcn_wmma_f32_16x16x4_f32(false, aMuB, false, bw1,
                                                                (short)0, accMuB1, false, false);
                accSgA0 = __builtin_amdgcn_wmma_f32_16x16x4_f32(false, aSgA, false, b20,
                                                                (short)0, accSgA0, false, false);
                accSgA1 = __builtin_amdgcn_wmma_f32_16x16x4_f32(false, aSgA, false, b21,
                                                                (short)0, accSgA1, false, false);
                accSgB0 = __builtin_amdgcn_wmma_f32_16x16x4_f32(false, aSgB, false, b20,
                                                                (short)0, accSgB0, false, false);
                accSgB1 = __builtin_amdgcn_wmma_f32_16x16x4_f32(false, aSgB, false, b21,
                                                                (short)0, accSgB1, false, false);

                rsA += aSgA.x + aSgA.y;
                rsA = fmaf(aMuA.x, aMuA.x, rsA);
                rsA = fmaf(aMuA.y, aMuA.y, rsA);
                rsB += aSgB.x + aSgB.y;
                rsB = fmaf(aMuB.x, aMuB.x, rsB);
                rsB = fmaf(aMuB.y, aMuB.y, rsB);
            }
        }
    }

    // Lane L and L+16 hold the K%4={0,1} and K%4={2,3} halves of row M=L&15.
    rsA += __shfl_xor(rsA, 16, 32);
    rsB += __shfl_xor(rsB, 16, 32);

    const float wsp0 = softplusf(w_sigma[nlo]) * inv_d;
    const float wsp1 = softplusf(w_sigma[16 + nlo]) * inv_d;

    float* outMu = out;
    float* outSg = out + OUTN;
    const int bpA = bA * PNUM;
    const int bpB = bB * PNUM;

    #pragma unroll
    for (int r = 0; r < 8; ++r) {
        const int Mr = r + laneHi * 8;                  // C/D layout row for VGPR r
        const float rsAm = __shfl(rsA, Mr, 32);
        const float rsBm = __shfl(rsB, Mr, 32);

        const int idxA0 = (bpA + pbA + Mr) * KNUM + nlo;
        const int idxA1 = idxA0 + 16;
        outMu[idxA0] = accMuA0[r];
        outMu[idxA1] = accMuA1[r];
        outSg[idxA0] = softplusf(accSgA0[r] + rsAm * wsp0);
        outSg[idxA1] = softplusf(accSgA1[r] + rsAm * wsp1);

        const int idxB0 = (bpB + pbB + Mr) * KNUM + nlo;
        const int idxB1 = idxB0 + 16;
        outMu[idxB0] = accMuB0[r];
        outMu[idxB1] = accMuB1[r];
        outSg[idxB0] = softplusf(accSgB0[r] + rsBm * wsp0);
        outSg[idxB1] = softplusf(accSgB1[r] + rsBm * wsp1);
    }
}

__global__ __launch_bounds__(256)
void vdp_conv_kl(const float* __restrict__ w_mu,
                 const float* __restrict__ w_sigma,
                 float* __restrict__ out)
{
    __shared__ float red[256];
    const int tid = threadIdx.x;
    float s = 0.0f;
    for (int i = tid; i < DK * KNUM; i += 256) {
        float w = w_mu[i];
        s = fmaf(w, w, s);
    }
    red[tid] = s;
    __syncthreads();
    for (int off = 128; off > 0; off >>= 1) {
        if (tid < off) red[tid] += red[tid + off];
        __syncthreads();
    }
    if (tid == 0) {
        const float sum_w2 = red[0];
        float sterm = 0.0f;
        for (int n = 0; n < KNUM; ++n) {
            float ws = w_sigma[n];
            sterm += -ws + softplusf(ws) / PRIOR_VAR;
        }
        float kl = 0.5f * (logf(PRIOR_VAR) - 1.0f
                           + sterm / (float)KNUM
                           + (sum_w2 / PRIOR_VAR) / (float)(DK * KNUM));
        out[2 * OUTN] = kl;
    }
}

extern "C" void kernel_launch(void* const* d_in, const int* in_sizes, int n_in,
                              void* d_out, int out_size, void* d_ws, size_t ws_size,
                              hipStream_t stream) {
    (void)in_sizes; (void)n_in; (void)out_size; (void)d_ws; (void)ws_size;
    const float* mu_in    = (const float*)d_in[0];
    const float* Sigma_in = (const float*)d_in[1];
    const float* w_mu     = (const float*)d_in[2];
    const float* w_sigma  = (const float*)d_in[3];
    float* out = (float*)d_out;

    const int total_pairs = (BSZ * TILES_PER_BATCH) / 2;      // 3600
    const int blocks = total_pairs / WAVES_PER_BLOCK;         // 450
    hipLaunchKernelGGL(vdp_conv_main, dim3(blocks), dim3(256), 0, stream,
                       mu_in, Sigma_in, w_mu, w_sigma, out);
    hipLaunchKernelGGL(vdp_conv_kl, dim3(1), dim3(256), 0, stream,
                       w_mu, w_sigma, out);
}